// DictionaryLearning_67491116089941
// MI455X (gfx1250) — compile-verified
//
#include <hip/hip_runtime.h>

// ---------------------------------------------------------------------------
// DictionaryLearning (batched OMP, K=5) for MI455X / gfx1250 (wave32, WMMA)
//
// d_in[0]: z_e        (64, 64, 32, 32) f32  -> signals X: (64 ch, 65536 sig)
// d_in[1]: dictionary (64, 512) f32
// d_out  : [ z_dl_st (4194304) | loss (1) | coeffs (512*65536) ]  f32
// d_ws   : [ d_norm (32768) | G (262144) | dnA (32768) ] f32  (~1.3 MB)
// ---------------------------------------------------------------------------

typedef __attribute__((ext_vector_type(2))) float v2f;
typedef __attribute__((ext_vector_type(8))) float v8f;
typedef unsigned int v4u __attribute__((ext_vector_type(4)));
typedef int          v4i __attribute__((ext_vector_type(4)));
typedef int          v8i __attribute__((ext_vector_type(8)));

#define NEMB 512
#define EDIM 64
#define KSP  5
#define BS   65536
#define HWSZ 1024
#define ZDL_N 4194304
#define EPSN 1e-10f
#define HB_PITCH 516   // 516 % 64 == 4 -> conflict-free across 16 sigs

// ---------------- dictionary column normalization ---------------------------
__global__ void dl_norm_kernel(const float* __restrict__ dict,
                               float* __restrict__ dn) {
  int n = blockIdx.x * blockDim.x + threadIdx.x;
  if (n >= NEMB) return;
  float s = 0.f;
  for (int c = 0; c < EDIM; ++c) { float v = dict[c * NEMB + n]; s += v * v; }
  float inv = 1.0f / fmaxf(sqrtf(s), EPSN);
  for (int c = 0; c < EDIM; ++c) dn[c * NEMB + n] = dict[c * NEMB + n] * inv;
}

// ---------------- repack dn into WMMA A/B-fragment order --------------------
// A (16x4 f32) lane layout: lane l holds A[m=l&15][k = 2*(l>>4) + v], v=0,1.
// dnA[((at*16 + c)*32 + l)] = float2{ dn[ch][at*16+mi], dn[ch+1][at*16+mi] },
//   ch = c*4 + 2*(l>>4), mi = l&15  ->  fragment load = coalesced b64.
__global__ void dl_repack_kernel(const float* __restrict__ dn,
                                 float* __restrict__ dnA) {
  const int t = blockIdx.x * blockDim.x + threadIdx.x;   // 0..16383
  const int l = t & 31, c = (t >> 5) & 15, at = t >> 9;
  const int mi = l & 15, hi = l >> 4;
  const int ch = c * 4 + 2 * hi;
  dnA[t * 2    ] = dn[(ch    ) * NEMB + at * 16 + mi];
  dnA[t * 2 + 1] = dn[(ch + 1) * NEMB + at * 16 + mi];
}

// ---------------- G = Dn^T Dn via f32 WMMA (16x16 tile / wave) --------------
// C (16x16 f32): VGPR i, lane l -> C[m = i + 8*(l>>4)][n = l&15]
__global__ void __launch_bounds__(32)
dl_gram_kernel(const float* __restrict__ dnA, float* __restrict__ G) {
  const int l = threadIdx.x;
  const int mi = l & 15, hi = l >> 4;
  const int bi = blockIdx.x, bj = blockIdx.y;
  const v2f* dA = (const v2f*)dnA;
  v8f acc = {};
#pragma unroll
  for (int c = 0; c < 16; ++c) {
    v2f a = dA[(bi * 16 + c) * 32 + l];
    v2f b = dA[(bj * 16 + c) * 32 + l];
    acc = __builtin_amdgcn_wmma_f32_16x16x4_f32(false, a, false, b,
                                                (short)0, acc, false, false);
  }
#pragma unroll
  for (int i = 0; i < 8; ++i)
    G[(bi * 16 + i + 8 * hi) * NEMB + bj * 16 + mi] = acc[i];
}

// ---------------- zero loss + dense coeffs region (134 MB) ------------------
// TDM path: zero one 16 KB LDS tile, then a single TENSOR_STORE_FROM_LDS per
// wave with iterate_enable=1, lds_addr_increment=0 streams 64 x 16 KB = 1 MB
// of zeros per instruction (D# per CDNA5 ISA 8.3-8.6). 128 blocks cover all
// 512*65536 coeffs. Falls back to a plain loop if the builtin is absent.
__global__ void __launch_bounds__(32)
dl_zero_kernel(float* __restrict__ out) {
  const int l = threadIdx.x;
  if (blockIdx.x == 0 && l == 0) out[ZDL_N] = 0.f;          // loss slot
  float* slice = out + ZDL_N + 1 + (size_t)blockIdx.x * 262144;  // 1 MB each
#if defined(__HIP_DEVICE_COMPILE__) && __has_builtin(__builtin_amdgcn_tensor_store_from_lds)
  __shared__ float4 zb[1024];                               // 16 KB of zeros
  float4 z; z.x = 0.f; z.y = 0.f; z.z = 0.f; z.w = 0.f;
  for (int i = l; i < 1024; i += 32) zb[i] = z;
  __syncthreads();
  const unsigned long long ga = (unsigned long long)slice;  // byte address
  v4u g0;
  g0[0] = 1u;                                   // count=1, user D#
  g0[1] = 0u;                                   // lds_addr = 0 (start of alloc)
  g0[2] = (unsigned)(ga & 0xFFFFFFFFull);       // global_addr[31:0]
  g0[3] = (unsigned)((ga >> 32) & 0x1FFFFFFull) | (2u << 30);  // [56:32]|type=2
  v8i g1;
  g1[0] = (2 << 16) | (1 << 19);  // data_size=4B, iterate_enable=1
  g1[1] = 0;                      // no atomic barrier; tensor_dim0[15:0]=0
  g1[2] = 0x4000;                 // tensor_dim0 = 0x40000000 (no OOB clip)
  g1[3] = (4096 << 16);           // tensor_dim1=0; tile_dim0 = 4096 elems
  g1[4] = 0;                      // tile_dim1/2 unused -> 1D tile
  g1[5] = 4096;                   // tensor_dim0_stride (unused for 1D)
  g1[6] = 0;
  g1[7] = 0;
  v4i g2;
  g2[0] = 1;                      // tensor_dim2 (unused)
  g2[1] = 0;                      // lds_addr_increment = 0 (reuse zero tile)
  g2[2] = 4096;                   // global_addr_increment = 16 KB (elems)
  g2[3] = (63 << 16);             // iterate_count: 64 iterations -> 1 MB
  v4i g3; g3[0] = 0; g3[1] = 0; g3[2] = 0; g3[3] = 0;
#if __clang_major__ >= 23
  v8i g4 = {};
  __builtin_amdgcn_tensor_store_from_lds(g0, g1, g2, g3, g4, 0);
#else
  __builtin_amdgcn_tensor_store_from_lds(g0, g1, g2, g3, 0);
#endif
  __builtin_amdgcn_s_wait_tensorcnt(0);
#else
  float4 z; z.x = 0.f; z.y = 0.f; z.z = 0.f; z.w = 0.f;
  float4* p = (float4*)slice;                  // slice is only 4B aligned (ZDL_N + 1 offset)
  for (int i = l; i < 65536; i += 32) p[i] = z;
#endif
}

// ---------------- main OMP kernel: 1 wave / block, 16 signals / wave --------
__global__ void __launch_bounds__(32)
dl_omp_kernel(const float* __restrict__ ze, const float* __restrict__ dnA,
              const float* __restrict__ dn, const float* __restrict__ G,
              float* __restrict__ zdl, float* __restrict__ loss,
              float* __restrict__ coeffs) {
  __shared__ float hb[16 * HB_PITCH];  // h_bar tile: [sig][atom], 33 KB

  const int l    = threadIdx.x;
  const int sig  = l & 15;             // signal within tile
  const int half = l >> 4;             // lane-pair half (atom/channel split)
  const int s_base = blockIdx.x * 16;
  const int b  = s_base / HWSZ;
  const int hw = s_base % HWSZ;        // 16 | 1024 -> tile never crosses batch
  const float* zbase = ze + (size_t)b * EDIM * HWSZ + hw;

  __builtin_prefetch(dnA + l * 64, 0, 3);   // global_prefetch_b8 warm-up

  // ---- B fragments are lane-private: load straight into registers --------
  // lane l needs X[c*4 + 2*half + v][sig], v=0,1 for each chunk c.
  v2f bfrag[16];
#pragma unroll
  for (int c = 0; c < 16; ++c) {
    const int ch = c * 4 + 2 * half;
    bfrag[c].x = zbase[(ch    ) * HWSZ + sig];
    bfrag[c].y = zbase[(ch + 1) * HWSZ + sig];
  }

  // ---- h_bar = Dn^T x via v_wmma_f32_16x16x4_f32; A streams as b64 --------
  const v2f* dA = (const v2f*)dnA;
  for (int at = 0; at < 32; ++at) {
    v8f acc = {};
#pragma unroll
    for (int c = 0; c < 16; ++c) {
      v2f a = dA[(at * 16 + c) * 32 + l];
      acc = __builtin_amdgcn_wmma_f32_16x16x4_f32(false, a, false, bfrag[c],
                                                  (short)0, acc, false, false);
    }
    // lane l holds atoms {at*16 + 8*half .. +7} of its sig: 2 x ds_store_b128
    float* dst = &hb[sig * HB_PITCH + at * 16 + 8 * half];
    float4 q0, q1;
    q0.x = acc[0]; q0.y = acc[1]; q0.z = acc[2]; q0.w = acc[3];
    q1.x = acc[4]; q1.y = acc[5]; q1.z = acc[6]; q1.w = acc[7];
    ((float4*)dst)[0] = q0;
    ((float4*)dst)[1] = q1;
  }
  __syncthreads();

  // ---- batched OMP, K=5; lane pair (l, l^16) serves signal `sig` ----------
  int   idx[KSP];
  float xv[KSP], hsel[KSP];
  float Lm[KSP][KSP];
  const float* hrow = &hb[sig * HB_PITCH];
  const int nbase = half * 256;

#pragma unroll
  for (int k = 0; k < KSP; ++k) {
    // argmax_n | h_bar[n] - (x.G)[n] | over this half's 256 atoms, 4 at a time
    float best = -1.f; int bestn = nbase;
    for (int q = 0; q < 64; ++q) {
      const int n0 = nbase + q * 4;
      const float4 hq = *(const float4*)(hrow + n0);      // ds_load_b128
      float hv[4] = {hq.x, hq.y, hq.z, hq.w};
#pragma unroll
      for (int j = 0; j < k; ++j) {
        const float4 g = *(const float4*)&G[idx[j] * NEMB + n0]; // b128 gather
        hv[0] -= xv[j] * g.x; hv[1] -= xv[j] * g.y;
        hv[2] -= xv[j] * g.z; hv[3] -= xv[j] * g.w;
      }
#pragma unroll
      for (int t = 0; t < 4; ++t) {
        const int n = n0 + t;
        bool sel = false;
#pragma unroll
        for (int j = 0; j < k; ++j) sel = sel || (n == idx[j]);
        const float a = sel ? 0.f : fabsf(hv[t]);
        if (a > best) { best = a; bestn = n; }
      }
    }
    // merge halves (first-max semantics: prefer lower index on tie)
    const float ob = __shfl_xor(best, 16, 32);
    const int   on = __shfl_xor(bestn, 16, 32);
    if (ob > best || (ob == best && on < bestn)) { best = ob; bestn = on; }
    idx[k] = bestn;

    // incremental Cholesky: append row k (redundant in both halves)
    if (k == 0) {
      Lm[0][0] = 1.f;
    } else {
      float w[KSP]; float ss = 0.f;
#pragma unroll
      for (int i = 0; i < k; ++i) {
        float t = G[idx[i] * NEMB + idx[k]];
#pragma unroll
        for (int j = 0; j < i; ++j) t -= Lm[i][j] * w[j];
        w[i] = t / Lm[i][i];
        ss += w[i] * w[i];
        Lm[k][i] = w[i];
      }
      Lm[k][k] = sqrtf(fmaxf(1.f - ss, 1e-20f));
    }

    hsel[k] = hrow[idx[k]];

    // solve L y = h_sel, then L^T x = y  (size k+1)
    float y[KSP];
#pragma unroll
    for (int i = 0; i <= k; ++i) {
      float t = hsel[i];
#pragma unroll
      for (int j = 0; j < i; ++j) t -= Lm[i][j] * y[j];
      y[i] = t / Lm[i][i];
    }
#pragma unroll
    for (int i = k; i >= 0; --i) {
      float t = y[i];
#pragma unroll
      for (int j = i + 1; j <= k; ++j) t -= Lm[j][i] * xv[j];
      xv[i] = t / Lm[i][i];
    }
  }

  // ---- scatter sparse coefficients (coeffs is (N, Bs) row-major) ----------
  if (half == 0) {
#pragma unroll
    for (int j = 0; j < KSP; ++j)
      coeffs[(size_t)idx[j] * BS + s_base + sig] = xv[j];
  }

  // ---- reconstruction, z_dl_st (== z_dl in value), loss partial -----------
  float lsum = 0.f;
  for (int c = half * 32; c < half * 32 + 32; ++c) {
    float r = 0.f;
#pragma unroll
    for (int j = 0; j < KSP; ++j) r += xv[j] * dn[c * NEMB + idx[j]];
    const float zev = zbase[c * HWSZ + sig];
    const float d = r - zev;
    lsum += d * d;
    zdl[(size_t)b * EDIM * HWSZ + c * HWSZ + hw + sig] = r;
  }

  // loss = dl_latent + 0.25*e_latent = 1.25 * mean((z_dl - z_e)^2)
  for (int off = 16; off; off >>= 1) lsum += __shfl_xor(lsum, off, 32);
  if (l == 0) atomicAdd(loss, lsum * (1.25f / (float)ZDL_N));
}

// ---------------------------------------------------------------------------
extern "C" void kernel_launch(void* const* d_in, const int* in_sizes, int n_in,
                              void* d_out, int out_size, void* d_ws, size_t ws_size,
                              hipStream_t stream) {
  const float* ze   = (const float*)d_in[0];
  const float* dict = (const float*)d_in[1];
  float* out = (float*)d_out;
  float* dn  = (float*)d_ws;            // 64*512 floats
  float* G   = dn + EDIM * NEMB;        // 512*512 floats
  float* dnA = G + NEMB * NEMB;         // 64*512 floats, fragment-packed
  float* zdl    = out;
  float* loss   = out + ZDL_N;
  float* coeffs = out + ZDL_N + 1;

  dl_norm_kernel<<<2, 256, 0, stream>>>(dict, dn);
  dl_repack_kernel<<<64, 256, 0, stream>>>(dn, dnA);
  dl_gram_kernel<<<dim3(32, 32), 32, 0, stream>>>(dnA, G);
  dl_zero_kernel<<<128, 32, 0, stream>>>(out);   // TDM bulk zero of coeffs
  dl_omp_kernel<<<BS / 16, 32, 0, stream>>>(ze, dnA, dn, G, zdl, loss, coeffs);
}